// ScanConversion2D_75136157876742
// MI455X (gfx1250) — compile-verified
//
#include <hip/hip_runtime.h>
#include <stdint.h>
#include <stddef.h>

// Problem geometry (fixed by reference: x = [B,1,1024,1024] fp32)
#define IMG_H 1024
#define IMG_W 1024
#define TILE_W 64
#define TILE_H 32
#define LROWS (TILE_H + 3)          // 35 rows:  r0-1 .. r0+TILE_H+1
#define LCOLS (TILE_W + 8)          // 72 cols:  c0-4 .. c0+TILE_W+3 (16B-aligned start)
#define SEGS  (LCOLS / 4)           // 18 x b128 segments per row
#define NSLOTS (LROWS * SEGS)       // 630 async b128 loads per tile

__device__ __forceinline__ float signf(float v) {
    // jnp.sign for finite inputs: -1 / 0 / +1
    return (float)((v > 0.0f) - (v < 0.0f));
}

__global__ __launch_bounds__(256)
void sobel_sign_fused(const float* __restrict__ x, float* __restrict__ out,
                      int B) {
    __shared__ __align__(16) float tile[LROWS][LCOLS];

    const int c0  = blockIdx.x * TILE_W;   // output tile origin (col)
    const int r0  = blockIdx.y * TILE_H;   // output tile origin (row)
    const int b   = blockIdx.z;            // batch
    const int tid = threadIdx.x;

    const float* xb = x + (size_t)b * IMG_H * IMG_W;
    // LDS byte address of tile[0][0] (wave-relative; single shared array -> base 0)
    const unsigned lds_base = (unsigned)(size_t)(void*)&tile[0][0];

    // ---- Stage (TILE_H+3) x (TILE_W+8) tile into LDS via async b128 loads ----
    for (int slot = tid; slot < NSLOTS; slot += 256) {
        const int row = slot / SEGS;
        const int seg = slot - row * SEGS;
        int gr = r0 - 1 + row;
        gr = gr < 0 ? 0 : (gr > IMG_H - 1 ? IMG_H - 1 : gr);   // clamp; zero-fixed later
        int gc = c0 - 4 + seg * 4;
        gc = gc < 0 ? 0 : (gc > IMG_W - 4 ? IMG_W - 4 : gc);   // stays 16B aligned
        const float* gp = xb + (size_t)gr * IMG_W + gc;
        const unsigned loff = lds_base + (unsigned)(row * (LCOLS * 4) + seg * 16);
        asm volatile("global_load_async_to_lds_b128 %0, %1, off"
                     :: "v"(loff), "v"(gp)
                     : "memory");
    }
    asm volatile("s_wait_asynccnt 0" ::: "memory");
    __syncthreads();

    // ---- Zero-fill out-of-image halo (SAME zero padding); boundary blocks only ----
    const bool boundary = (r0 == 0) || (r0 + TILE_H + 1 >= IMG_H) ||
                          (c0 == 0) || (c0 + TILE_W + 3 >= IMG_W);
    if (boundary) {
        for (int s = tid; s < LROWS * LCOLS; s += 256) {
            const int row = s / LCOLS;
            const int col = s - row * LCOLS;
            const int gr = r0 - 1 + row;
            const int gc = c0 - 4 + col;
            if (gr < 0 || gr >= IMG_H || gc < 0 || gc >= IMG_W)
                tile[row][col] = 0.0f;
        }
        __syncthreads();
    }

    // ---- Compute: each thread does 8 pixels of a 64x32 tile ----
    const int tx  = tid & 63;        // 0..63 (column within tile)
    const int tyb = tid >> 6;        // 0..3  (row phase)
    const int gj  = c0 + tx;
    const bool jok = (gj < IMG_W - 1);

    const size_t Y0_BASE = (size_t)B * 2 * IMG_H * IMG_W;
    const size_t Y1_BASE = Y0_BASE + (size_t)B * (IMG_H - 1) * (IMG_W - 1);

    #pragma unroll
    for (int k = 0; k < 8; ++k) {
        const int iy = tyb + k * 4;      // 0..31
        const int gi = r0 + iy;
        const int cs = tx + 3;           // LDS col of (gi, gj-1)

        // 4x4 neighborhood: rows gi-1..gi+2, cols gj-1..gj+2
        const float a00 = tile[iy + 0][cs + 0], a01 = tile[iy + 0][cs + 1],
                    a02 = tile[iy + 0][cs + 2], a03 = tile[iy + 0][cs + 3];
        const float a10 = tile[iy + 1][cs + 0], a11 = tile[iy + 1][cs + 1],
                    a12 = tile[iy + 1][cs + 2], a13 = tile[iy + 1][cs + 3];
        const float a20 = tile[iy + 2][cs + 0], a21 = tile[iy + 2][cs + 1],
                    a22 = tile[iy + 2][cs + 2], a23 = tile[iy + 2][cs + 3];
        const float a30 = tile[iy + 3][cs + 0], a31 = tile[iy + 3][cs + 1],
                    a32 = tile[iy + 3][cs + 2];

        // Cross-correlation with Gx=[[1,0,-1],[2,0,-2],[1,0,-1]], Gy=[[1,2,1],[0,0,0],[-1,-2,-1]]
        const float gx  = (a00 - a02) + 2.0f * (a10 - a12) + (a20 - a22);
        const float gxr = (a01 - a03) + 2.0f * (a11 - a13) + (a21 - a23);  // gx at (gi, gj+1)
        const float gy  = (a00 + 2.0f * a01 + a02) - (a20 + 2.0f * a21 + a22);
        const float gyd = (a10 + 2.0f * a11 + a12) - (a30 + 2.0f * a31 + a32); // gy at (gi+1, gj)

        // y[b,0], y[b,1]  (streamed -> non-temporal)
        const size_t o0 = (((size_t)(b * 2 + 0) * IMG_H) + gi) * IMG_W + gj;
        const size_t o1 = (((size_t)(b * 2 + 1) * IMG_H) + gi) * IMG_W + gj;
        __builtin_nontemporal_store(gx, &out[o0]);
        __builtin_nontemporal_store(gy, &out[o1]);

        // y0/y1: [B,1,1023,1023]
        if (jok && gi < IMG_H - 1) {
            const size_t e = ((size_t)b * (IMG_H - 1) + gi) * (IMG_W - 1) + gj;
            __builtin_nontemporal_store(signf(gx) - signf(gxr), &out[Y0_BASE + e]);
            __builtin_nontemporal_store(signf(gy) - signf(gyd), &out[Y1_BASE + e]);
        }
    }
}

extern "C" void kernel_launch(void* const* d_in, const int* in_sizes, int n_in,
                              void* d_out, int out_size, void* d_ws, size_t ws_size,
                              hipStream_t stream) {
    (void)n_in; (void)out_size; (void)d_ws; (void)ws_size;
    const float* x = (const float*)d_in[0];
    float* out = (float*)d_out;
    const int B = in_sizes[0] / (IMG_H * IMG_W);   // 16 for the reference shapes

    dim3 grid(IMG_W / TILE_W, IMG_H / TILE_H, B);  // 16 x 32 x B blocks
    sobel_sign_fused<<<grid, 256, 0, stream>>>(x, out, B);
}